// VectorQuantizer_89635967468152
// MI455X (gfx1250) — compile-verified
//
#include <hip/hip_runtime.h>

typedef __attribute__((ext_vector_type(2))) float v2f;
typedef __attribute__((ext_vector_type(8))) float v8f;

#define N_CODES        1024
#define CODE_DIM       64
#define ROWS_PER_BLOCK 256     // 8 waves x 2 M-tiles x 16 rows
#define THREADS        256
#define NT_COUNT       (N_CODES / 16)

// LDS layout (bytes):
//   s_b   : [1024][34] float2  @ 0        (278528)  B fragments, pad 34 -> conflict-free b64
//   s_wn  : [1024]     f32     @ 278528   (  4096)
//   s_idx : [256]      i32     @ 282624   (  1024)
// total = 283648   (CDNA5: 320KB per-workgroup LDS)
#define SB_OFF     0
#define WN_OFF     278528
#define IDX_OFF    282624
#define SMEM_BYTES 283648
#define SB_STRIDE  34      // float2 units per code

// Load the 16 B fragments of N-tile NT into register array ARR (conflict-free b64s).
#define LOADB(NT, ARR)                                        \
    {                                                         \
        const float2* bp = bbase + (NT) * (16 * SB_STRIDE);   \
        _Pragma("unroll")                                     \
        for (int s = 0; s < 16; ++s) {                        \
            float2 t = bp[2 * s];                             \
            ARR[s].x = t.x;                                   \
            ARR[s].y = t.y;                                   \
        }                                                     \
    }

// Two independent 16-WMMA accumulation chains (M-tiles 0/1) + argmin update.
#define WMMA_STEP(BARR, NT)                                                     \
    {                                                                           \
        const int nn = (NT) * 16 + ncol;                                        \
        const float wn = s_wn[nn];                                              \
        v8f a0 = {}, a1 = {};                                                   \
        _Pragma("unroll")                                                       \
        for (int s = 0; s < 16; ++s)                                            \
            a0 = __builtin_amdgcn_wmma_f32_16x16x4_f32(                         \
                false, af0[s], false, BARR[s], (short)0, a0, false, false);     \
        _Pragma("unroll")                                                       \
        for (int s = 0; s < 16; ++s)                                            \
            a1 = __builtin_amdgcn_wmma_f32_16x16x4_f32(                         \
                false, af1[s], false, BARR[s], (short)0, a1, false, false);     \
        _Pragma("unroll")                                                       \
        for (int v = 0; v < 8; ++v) {                                           \
            float d0 = __builtin_fmaf(-2.0f, a0[v], wn);                        \
            float d1 = __builtin_fmaf(-2.0f, a1[v], wn);                        \
            if (d0 < b0d[v]) { b0d[v] = d0; b0i[v] = nn; }                      \
            if (d1 < b1d[v]) { b1d[v] = d1; b1i[v] = nn; }                      \
        }                                                                       \
    }

__global__ __launch_bounds__(THREADS) void vq_wmma_kernel(
    const float* __restrict__ X,    // [32,64,64,64]  (b, c, h, w)
    const float* __restrict__ CB,   // [1024,64]
    float* __restrict__ OUT)        // [32,64,64,64]  (b, h, w, c) flat
{
    extern __shared__ __align__(16) char smem[];
    float2* s_b  = (float2*)(smem + SB_OFF);
    float*  s_wn = (float*) (smem + WN_OFF);
    int*    s_idx= (int*)   (smem + IDX_OFF);

    const int tid  = threadIdx.x;
    const int lane = tid & 31;
    const int wave = tid >> 5;

    // ---- stage codebook into LDS pre-swizzled into B-fragment layout; norms for free ----
    // fragment (s, h) of code n: {cb[n][4s+h], cb[n][4s+2+h]}  at  s_b[n*34 + 2s + h]
    {
        const float4* cb4 = (const float4*)CB;   // [1024][16] float4
        #pragma unroll
        for (int i = 0; i < N_CODES / THREADS; ++i) {
            const int n = i * THREADS + tid;
            float2* bn = s_b + n * SB_STRIDE;
            float nrm = 0.f;
            #pragma unroll
            for (int s = 0; s < 16; ++s) {
                float4 q = cb4[n * 16 + s];      // cb[n][4s .. 4s+3]
                nrm += q.x * q.x + q.y * q.y + q.z * q.z + q.w * q.w;
                bn[2 * s + 0] = make_float2(q.x, q.z);   // h = 0
                bn[2 * s + 1] = make_float2(q.y, q.w);   // h = 1
            }
            s_wn[n] = nrm;
        }
    }

    // ---- A fragments straight from global (coalesced: 16 consecutive floats per half-wave) ----
    const int p0  = blockIdx.x * ROWS_PER_BLOCK + wave * 32;  // first of this wave's 32 rows
    const int bb  = p0 >> 12;                                 // 4096 points per image
    const int hw0 = p0 & 4095;
    const float* xw = X + (size_t)bb * (CODE_DIM * 4096) + hw0;
    const int rsel = lane & 15;
    const int ksel = (lane < 16) ? 0 : 2;

    v2f af0[16], af1[16];                 // M-tile 0 (rows 0-15), M-tile 1 (rows 16-31)
    #pragma unroll
    for (int s = 0; s < 16; ++s) {
        const float* p = xw + (4 * s + ksel) * 4096 + rsel;
        af0[s].x = p[0];
        af0[s].y = p[4096];
        af1[s].x = p[16];
        af1[s].y = p[4096 + 16];
    }

    __syncthreads();                      // codebook fragments + norms visible

    float b0d[8], b1d[8];
    int   b0i[8], b1i[8];
    #pragma unroll
    for (int v = 0; v < 8; ++v) {
        b0d[v] = 3.0e38f; b0i[v] = 0;
        b1d[v] = 3.0e38f; b1i[v] = 0;
    }

    const int ncol = lane & 15;           // C/D column owned by this lane
    const int hsel = lane >> 4;           // B fragment half select
    const float2* bbase = s_b + ncol * SB_STRIDE + hsel;

    // ---- pipelined sweep: prefetch next tile's B while this tile's 32 WMMAs run ----
    v2f bufA[16], bufB[16];
    LOADB(0, bufA);
    for (int nt = 0; nt < NT_COUNT; nt += 2) {
        LOADB(nt + 1, bufB);
        WMMA_STEP(bufA, nt);
        LOADB(nt + 2, bufA);   // final prefetch reads in-bounds LDS garbage (s_wn), unused
        WMMA_STEP(bufB, nt + 1);
    }

    // ---- argmin across the 16 lanes of each half (wave32 butterfly) ----
    #pragma unroll
    for (int m = 1; m <= 8; m <<= 1) {
        #pragma unroll
        for (int v = 0; v < 8; ++v) {
            float od0 = __shfl_xor(b0d[v], m, 32);
            int   oi0 = __shfl_xor(b0i[v], m, 32);
            if (od0 < b0d[v] || (od0 == b0d[v] && oi0 < b0i[v])) { b0d[v] = od0; b0i[v] = oi0; }
            float od1 = __shfl_xor(b1d[v], m, 32);
            int   oi1 = __shfl_xor(b1i[v], m, 32);
            if (od1 < b1d[v] || (od1 == b1d[v] && oi1 < b1i[v])) { b1d[v] = od1; b1i[v] = oi1; }
        }
    }

    // lanes 0-15 hold rows r=v; lanes 16-31 hold rows r=v+8 (per M-tile)
    if (lane == 0 || lane == 16) {
        int rbase = wave * 32 + ((lane == 16) ? 8 : 0);
        #pragma unroll
        for (int v = 0; v < 8; ++v) {
            s_idx[rbase + v]      = b0i[v];
            s_idx[rbase + 16 + v] = b1i[v];
        }
    }
    __syncthreads();

    // ---- gather codebook rows (L2-resident) -> coalesced [b,h,w,c] output ----
    float* outb = OUT + (size_t)blockIdx.x * (ROWS_PER_BLOCK * CODE_DIM);
    #pragma unroll 4
    for (int j = 0; j < ROWS_PER_BLOCK * CODE_DIM / THREADS; ++j) {
        int flat = j * THREADS + tid;
        int row  = flat >> 6;
        int c    = flat & 63;
        outb[flat] = CB[s_idx[row] * CODE_DIM + c];
    }
}

extern "C" void kernel_launch(void* const* d_in, const int* in_sizes, int n_in,
                              void* d_out, int out_size, void* d_ws, size_t ws_size,
                              hipStream_t stream) {
    const float* X  = (const float*)d_in[0];   // vectors  [32,64,64,64] f32
    const float* CB = (const float*)d_in[1];   // codebook [1024,64]     f32
    float* OUT = (float*)d_out;

    const int npoints = in_sizes[0] / CODE_DIM;        // 131072
    const int blocks  = npoints / ROWS_PER_BLOCK;      // 512

    // Opt in to >64KB dynamic LDS (CDNA5: 320KB per workgroup).
    hipFuncSetAttribute((const void*)vq_wmma_kernel,
                        hipFuncAttributeMaxDynamicSharedMemorySize, SMEM_BYTES);

    hipLaunchKernelGGL(vq_wmma_kernel, dim3(blocks), dim3(THREADS), SMEM_BYTES,
                       stream, X, CB, OUT);
}